// IFConstantCurrentEncoder_68264210203029
// MI455X (gfx1250) — compile-verified
//
#include <hip/hip_runtime.h>

// IF constant-current encoder for MI455X (gfx1250, wave32).
//
// Store-bandwidth bound: 512 MB output @ 23.3 TB/s => ~22 us floor.
// Optimal shape:
//   - lane owns 4 consecutive neurons -> two global_store_b128 (non-temporal;
//     512 MB stream >> 192 MB L2) per timestep
//   - packed-f32 math: v_pk_add_f32 / v_floor_f32 / v_pk_fma_f32
//     (z = floor(v) == (v>=1) since v in [0,2); v -= z*v == reset-to-zero,
//     bitwise identical to reference's (1-z)*v for z in {0,1})
//   - 32768 float4-threads = 1024 waves, 64-thread blocks spread across WGPs

typedef float v4f __attribute__((ext_vector_type(4)));

__global__ __launch_bounds__(64, 1)
void IFConstantCurrentEncoder_kernel(const float* __restrict__ x,
                                     float* __restrict__ spikes,
                                     float* __restrict__ volts,
                                     int n_total,   // B*N flat neurons (131072)
                                     int T)         // seq_length (512)
{
    const int i4   = blockIdx.x * blockDim.x + threadIdx.x; // float4 index
    const int base = i4 * 4;
    if (base + 3 >= n_total) return;                        // n_total % 4 == 0

    // dv = DT * TAU_MEM_INV * x = 0.1f * x   (v_pk_mul_f32)
    const v4f xv = *(const v4f*)(x + base);
    const v4f dv = xv * 0.1f;
    v4f v = {0.0f, 0.0f, 0.0f, 0.0f};

    const size_t stride = (size_t)n_total;
    float* __restrict__ sp = spikes + base;
    float* __restrict__ vp = volts  + base;
    size_t off = 0;

    #pragma unroll 8
    for (int t = 0; t < T; ++t) {
        v = v + dv;                                    // 2x v_pk_add_f32
        const v4f z = __builtin_elementwise_floor(v);  // 4x v_floor_f32 ; spike flag
        v = v - z * v;                                 // 2x v_pk_fma_f32 ; reset on spike

        // Streaming 128-bit non-temporal stores
        __builtin_nontemporal_store(z, (v4f*)(sp + off));
        __builtin_nontemporal_store(v, (v4f*)(vp + off));
        off += stride;
    }
}

extern "C" void kernel_launch(void* const* d_in, const int* in_sizes, int n_in,
                              void* d_out, int out_size, void* d_ws, size_t ws_size,
                              hipStream_t stream) {
    (void)n_in; (void)d_ws; (void)ws_size;

    const float* x = (const float*)d_in[0];       // [B, N] fp32
    const int n_total = in_sizes[0];              // B*N = 131072

    // d_out = concat(spikes[T,B,N], voltages[T,B,N]) flat.
    const int half = out_size / 2;                // T * n_total
    const int T    = half / n_total;              // 512

    float* spikes = (float*)d_out;
    float* volts  = (float*)d_out + (size_t)half;

    const int num4  = n_total / 4;                // 32768 float4-threads
    const int block = 64;                         // 2 waves/WG, spread over WGPs
    const int grid  = (num4 + block - 1) / block;

    IFConstantCurrentEncoder_kernel<<<grid, block, 0, stream>>>(
        x, spikes, volts, n_total, T);
}